// ChamferDistance_42082089566508
// MI455X (gfx1250) — compile-verified
//
#include <hip/hip_runtime.h>
#include <hip/hip_bf16.h>
#include <float.h>

// ---------------------------------------------------------------------------
// Chamfer distance (squared), B=8, N=M=8192, D=3, fp32.
// dist(x,y) = |x|^2 + |y|^2 - 2 x.y  computed via V_WMMA_F32_16X16X4_F32:
//   A row (per x-point):  [-2x0, -2x1, -2x2, |x|^2]   (K=4, D=3 padded)
//   B col (per y-point):  [ y0,   y1,   y2,   1    ]
//   C = 0 (inline, loop-invariant)
// => D[n][m] = |x_n|^2 - 2 x_n.y_m ; epilogue adds per-lane |y_m|^2 and mins.
// ---------------------------------------------------------------------------

typedef __attribute__((ext_vector_type(2))) float v2f;
typedef __attribute__((ext_vector_type(8))) float v8f;

// One wave (32 threads) per block. Each wave owns a 16-row strip of P and
// scans all columns of Q, producing rowmin[row] = min_j dist(P[row], Q[j]).
__global__ __launch_bounds__(32)
void chamfer_rowmin_wmma(const float* __restrict__ P,   // [B, nRows, 3]
                         const float* __restrict__ Q,   // [B, nCols, 3]
                         float* __restrict__ rowmin,    // [B, nRows]
                         int nRows, int nCols) {
  const int rowTiles = nRows >> 4;
  const int b  = blockIdx.x / rowTiles;
  const int rt = blockIdx.x % rowTiles;

  const int lane = threadIdx.x & 31;
  const int half = lane >> 4;     // 0: K=0/1 slice, 1: K=2/3 slice
  const int lidx = lane & 15;     // A: M index; B: N index

  const float* __restrict__ pb = P + (size_t)b * nRows * 3;
  const float* __restrict__ qb = Q + (size_t)b * nCols * 3;

  // ---- Load A strip: 16 x-points, layout per ISA 32-bit A 16x4 ----
  // VGPR0: lanes0-15 K=0, lanes16-31 K=2 ; VGPR1: lanes0-15 K=1, lanes16-31 K=3
  const int row = rt * 16 + lidx;
  const float x0 = pb[(size_t)row * 3 + 0];
  const float x1 = pb[(size_t)row * 3 + 1];
  const float x2 = pb[(size_t)row * 3 + 2];
  const float X2 = x0 * x0 + x1 * x1 + x2 * x2;

  v2f a;
  a.x = half ? (-2.0f * x2) : (-2.0f * x0);   // K=2 : K=0
  a.y = half ? X2           : (-2.0f * x1);   // K=3 : K=1

  v8f czero;
#pragma unroll
  for (int r = 0; r < 8; ++r) czero[r] = 0.0f;   // loop-invariant accumulator

  float rm[8];
#pragma unroll
  for (int r = 0; r < 8; ++r) rm[r] = FLT_MAX;

  const int colTiles = nCols >> 4;
#pragma unroll 2
  for (int jt = 0; jt < colTiles; ++jt) {
    // ---- Load B tile: 16 y-points, layout mirrors A (32-bit B 4x16) ----
    const int col = jt * 16 + lidx;
    const float y0 = qb[(size_t)col * 3 + 0];
    const float y1 = qb[(size_t)col * 3 + 1];
    const float y2 = qb[(size_t)col * 3 + 2];
    const float Y2 = y0 * y0 + y1 * y1 + y2 * y2;

    v2f bm;
    bm.x = half ? y2   : y0;    // K=2 : K=0
    bm.y = half ? 1.0f : y1;    // K=3 : K=1

    // S[n][m] = |x_n|^2 - 2 x_n . y_m
    v8f d = __builtin_amdgcn_wmma_f32_16x16x4_f32(
        /*neg_a=*/false, a, /*neg_b=*/false, bm,
        /*c_mod=*/(short)0, czero, /*reuse_a=*/false, /*reuse_b=*/false);

    // dist = S + |y_m|^2 ; Y2 is constant per lane (N = lane%16)
#pragma unroll
    for (int r = 0; r < 8; ++r) rm[r] = fminf(rm[r], d[r] + Y2);
  }

  // ---- Min-reduce across the 16 lanes of each half (wave32 shuffles) ----
#pragma unroll
  for (int r = 0; r < 8; ++r) {
    float v = rm[r];
    v = fminf(v, __shfl_xor(v, 1, 32));
    v = fminf(v, __shfl_xor(v, 2, 32));
    v = fminf(v, __shfl_xor(v, 4, 32));
    v = fminf(v, __shfl_xor(v, 8, 32));
    rm[r] = v;
  }

  // C/D layout: VGPR r holds row r (lanes 0-15) and row r+8 (lanes 16-31).
  float* __restrict__ out = rowmin + (size_t)b * nRows + rt * 16;
  if (lane == 0) {
#pragma unroll
    for (int r = 0; r < 8; ++r) out[r] = rm[r];
  }
  if (lane == 16) {
#pragma unroll
    for (int r = 0; r < 8; ++r) out[8 + r] = rm[r];
  }
}

// Per-batch mean of both row-min arrays; out[b] = mean(rmX[b]) + mean(rmY[b]).
__global__ __launch_bounds__(256)
void chamfer_reduce(const float* __restrict__ rowminX,
                    const float* __restrict__ rowminY,
                    float* __restrict__ out, int N, int M) {
  __shared__ float sdata[256];
  const int b = blockIdx.x;
  const float* rx = rowminX + (size_t)b * N;
  const float* ry = rowminY + (size_t)b * M;

  float sx = 0.0f, sy = 0.0f;
  for (int i = threadIdx.x; i < N; i += blockDim.x) sx += rx[i];
  for (int i = threadIdx.x; i < M; i += blockDim.x) sy += ry[i];

  sdata[threadIdx.x] = sx / (float)N + sy / (float)M;
  __syncthreads();
  for (int off = 128; off > 0; off >>= 1) {
    if ((int)threadIdx.x < off) sdata[threadIdx.x] += sdata[threadIdx.x + off];
    __syncthreads();
  }
  if (threadIdx.x == 0) out[b] = sdata[0];
}

extern "C" void kernel_launch(void* const* d_in, const int* in_sizes, int n_in,
                              void* d_out, int out_size, void* d_ws, size_t ws_size,
                              hipStream_t stream) {
  const float* X = (const float*)d_in[0];   // [B, N, 3]
  const float* Y = (const float*)d_in[1];   // [B, M, 3]
  float* out = (float*)d_out;               // [B]

  const int B = 8, D = 3;
  const int N = in_sizes[0] / (B * D);      // 8192
  const int M = in_sizes[1] / (B * D);      // 8192

  float* rowminX = (float*)d_ws;                    // B*N floats
  float* rowminY = rowminX + (size_t)B * N;         // B*M floats

  // Pass 1: min over Y for each X row.  Pass 2: roles swapped (symmetric).
  chamfer_rowmin_wmma<<<B * (N / 16), 32, 0, stream>>>(X, Y, rowminX, N, M);
  chamfer_rowmin_wmma<<<B * (M / 16), 32, 0, stream>>>(Y, X, rowminY, M, N);
  chamfer_reduce<<<B, 256, 0, stream>>>(rowminX, rowminY, out, N, M);
}